// Wrapper_21388937134385
// MI455X (gfx1250) — compile-verified
//
#include <hip/hip_runtime.h>
#include <cstdint>
#include <cstddef>

#define NBOX 8400
#define NCLS 80
#define ROWF 85            // 5 + 80
#define NW   263           // ceil(NBOX/32)
#define CONF_THRE 0.01f
#define NMS_THRE  0.2f

// ---------------------------------------------------------------------------
// CDNA5 async global -> LDS copy (tracked by ASYNCcnt), per ISA §15.18.3 /
// CDNA5 ISA. VDST operand is the LDS byte address (low 32
// bits of the generic LDS pointer), VADDR is the 64-bit global address.
// ---------------------------------------------------------------------------
__device__ __forceinline__ void async_copy16(void* lds_dst, const void* gsrc) {
  unsigned ldsoff = (unsigned)(size_t)lds_dst;   // LDS aperture: addr[31:0]
  asm volatile("global_load_async_to_lds_b128 %0, %1, off"
               :: "v"(ldsoff), "v"(gsrc) : "memory");
}
__device__ __forceinline__ void wait_async() {
  asm volatile("s_wait_asynccnt 0" ::: "memory");
}

// ---------------------------------------------------------------------------
// Kernel 1: decode boxes, class max/argmax (one wave32 per row), validity.
// ---------------------------------------------------------------------------
__global__ void k_prep(const float* __restrict__ pred,
                       float* __restrict__ boxes, float* __restrict__ det,
                       float* __restrict__ score, int* __restrict__ valid) {
  int wave = threadIdx.x >> 5;
  int lane = threadIdx.x & 31;
  int row  = blockIdx.x * 8 + wave;
  if (row >= NBOX) return;
  const float* p = pred + (size_t)row * ROWF;
  float cx = p[0], cy = p[1], w = p[2], h = p[3], obj = p[4];

  float best = -__builtin_inff();
  int   bidx = 0;
  for (int c = lane; c < NCLS; c += 32) {
    float v = p[5 + c];
    if (v > best) { best = v; bidx = c; }     // ascending c => first index
  }
  // butterfly reduction; ties -> smaller class index (jnp.argmax semantics)
  for (int off = 16; off > 0; off >>= 1) {
    float ob = __shfl_xor(best, off, 32);
    int   oi = __shfl_xor(bidx, off, 32);
    if (ob > best || (ob == best && oi < bidx)) { best = ob; bidx = oi; }
  }
  if (lane == 0) {
    float x1 = cx - w * 0.5f, y1 = cy - h * 0.5f;
    float x2 = cx + w * 0.5f, y2 = cy + h * 0.5f;
    boxes[(size_t)row * 4 + 0] = x1;
    boxes[(size_t)row * 4 + 1] = y1;
    boxes[(size_t)row * 4 + 2] = x2;
    boxes[(size_t)row * 4 + 3] = y2;
    float* d = det + (size_t)row * 7;
    d[0] = x1; d[1] = y1; d[2] = x2; d[3] = y2;
    d[4] = obj; d[5] = best; d[6] = (float)bidx;
    int v = (obj >= CONF_THRE) ? 1 : 0;
    valid[row] = v;
    score[row] = v ? obj : -1.0f;
  }
}

// ---------------------------------------------------------------------------
// Kernel 2: stable descending rank (== inverse of argsort(-scores)) against
// the full LDS-resident score vector (33.6 KB, async-staged), then scatter
// sorted boxes/valid.
// ---------------------------------------------------------------------------
__global__ void k_rank(const float* __restrict__ score,
                       const float* __restrict__ boxes,
                       const int* __restrict__ valid,
                       int* __restrict__ rank,
                       float* __restrict__ boxes_s, int* __restrict__ valid_s) {
  __shared__ __align__(16) float s_score[NBOX];
  for (int q = threadIdx.x; q < NBOX / 4; q += blockDim.x)   // 2100 x 16B
    async_copy16(&s_score[q * 4], &score[q * 4]);
  wait_async();
  __syncthreads();

  int i = blockIdx.x * blockDim.x + threadIdx.x;
  if (i < NBOX) {
    float si = s_score[i];
    int cnt = 0;
    for (int j = 0; j < NBOX; ++j) {
      float sj = s_score[j];
      cnt += (sj > si) || (sj == si && j < i);
    }
    rank[i] = cnt;
    float4 b = ((const float4*)boxes)[i];
    ((float4*)boxes_s)[cnt] = b;
    valid_s[cnt] = valid[i];
  }
}

// ---------------------------------------------------------------------------
// Kernel 3: suppression bit-matrix. Block = 8 rows x 32 col-words; the 1024
// column boxes (16 KB) are async-staged into LDS. Each thread emits one
// 32-bit word = 32 IoU tests. IEEE order matches the reference; contraction
// disabled so iou>0.2 comparisons are bit-faithful.
// ---------------------------------------------------------------------------
__global__ void k_mask(const float* __restrict__ boxes_s,
                       const int* __restrict__ valid_s,
                       unsigned* __restrict__ mask) {
#pragma clang fp contract(off)
  __shared__ __align__(16) float s_jb[1024 * 4];
  int jbase = blockIdx.x * 1024;
  for (int q = threadIdx.x; q < 1024; q += 256) {
    int j = jbase + q;
    if (j > NBOX - 1) j = NBOX - 1;          // clamp (16B-aligned, in-bounds)
    async_copy16(&s_jb[q * 4], &boxes_s[(size_t)j * 4]);
  }
  wait_async();
  __syncthreads();

  int tx = threadIdx.x & 31;                 // col-word within tile
  int ty = threadIdx.x >> 5;                 // row within tile
  int i  = blockIdx.y * 8 + ty;
  int cw = blockIdx.x * 32 + tx;
  if (i >= NBOX || cw >= NW) return;

  unsigned word = 0u;
  if (valid_s[i]) {                          // only valid rows can suppress
    float4 bi = ((const float4*)boxes_s)[i];
    float area_i = (bi.z - bi.x) * (bi.w - bi.y);
    for (int b = 0; b < 32; ++b) {
      int j = cw * 32 + b;
      if (j <= i || j >= NBOX) continue;     // strictly j > i, in range
      int q = tx * 32 + b;
      float jx1 = s_jb[q * 4 + 0], jy1 = s_jb[q * 4 + 1];
      float jx2 = s_jb[q * 4 + 2], jy2 = s_jb[q * 4 + 3];
      float area_j = (jx2 - jx1) * (jy2 - jy1);
      float ltx = fmaxf(bi.x, jx1), lty = fmaxf(bi.y, jy1);
      float rbx = fminf(bi.z, jx2), rby = fminf(bi.w, jy2);
      float wx = fmaxf(rbx - ltx, 0.0f), wy = fmaxf(rby - lty, 0.0f);
      float inter = wx * wy;
      float den = area_i + area_j - inter + 1e-9f;
      float iou = inter / den;
      if (iou > NMS_THRE) word |= (1u << b);
    }
  }
  mask[(size_t)i * NW + cw] = word;
}

// ---------------------------------------------------------------------------
// Kernel 4: greedy collect (the only serial part). One workgroup, keep-state
// in LDS (263 words). Per 32-row chunk: serial intra-word resolve, then
// parallel OR of the alive rows' mask words over remaining columns
// (L2-resident 8.8 MB matrix).
// ---------------------------------------------------------------------------
__global__ void k_collect(const unsigned* __restrict__ mask,
                          unsigned* __restrict__ removed_out) {
  __shared__ unsigned s_rm[NW];
  __shared__ unsigned s_diag[32];
  __shared__ unsigned s_alive;
  for (int c = threadIdx.x; c < NW; c += blockDim.x) s_rm[c] = 0u;
  __syncthreads();

  for (int c = 0; c < NW; ++c) {
    if (threadIdx.x < 32) {
      int i = c * 32 + (int)threadIdx.x;
      s_diag[threadIdx.x] = (i < NBOX) ? mask[(size_t)i * NW + c] : 0u;
    }
    __syncthreads();
    if (threadIdx.x == 0) {
      unsigned w = s_rm[c];
      for (int b = 0; b < 32; ++b) {
        int i = c * 32 + b;
        if (i < NBOX && !((w >> b) & 1u)) w |= s_diag[b];  // bits>b only
      }
      s_rm[c]  = w;
      s_alive  = ~w;
    }
    __syncthreads();
    unsigned alive = s_alive;
    for (int col = c + 1 + (int)threadIdx.x; col < NW; col += blockDim.x) {
      unsigned acc = 0u, am = alive;
      while (am) {
        int b = __builtin_ctz(am);
        am &= am - 1u;
        int i = c * 32 + b;
        if (i < NBOX) acc |= mask[(size_t)i * NW + col];
      }
      s_rm[col] |= acc;                      // each col owned by one thread
    }
    __syncthreads();
  }
  for (int c = threadIdx.x; c < NW; c += blockDim.x) removed_out[c] = s_rm[c];
}

// ---------------------------------------------------------------------------
// Kernel 5: out[rank[i]] = det[i] * (kept & valid)
// ---------------------------------------------------------------------------
__global__ void k_final(const float* __restrict__ det,
                        const int* __restrict__ rank,
                        const int* __restrict__ valid,
                        const unsigned* __restrict__ removed,
                        float* __restrict__ out) {
  int i = blockIdx.x * blockDim.x + threadIdx.x;
  if (i >= NBOX) return;
  int p = rank[i];
  bool keep = (((removed[p >> 5] >> (p & 31)) & 1u) == 0u) && (valid[i] != 0);
  float f = keep ? 1.0f : 0.0f;
  const float* d = det + (size_t)i * 7;
  float* o = out + (size_t)p * 7;
  for (int k = 0; k < 7; ++k) o[k] = d[k] * f;
}

// ---------------------------------------------------------------------------
extern "C" void kernel_launch(void* const* d_in, const int* in_sizes, int n_in,
                              void* d_out, int out_size, void* d_ws, size_t ws_size,
                              hipStream_t stream) {
  (void)in_sizes; (void)n_in; (void)out_size; (void)ws_size;
  const float* pred = (const float*)d_in[0];
  float* out = (float*)d_out;

  char* w = (char*)d_ws;                     // all offsets 16B-aligned
  float*    boxes   = (float*)   (w + 0);          // 134400 B
  float*    det     = (float*)   (w + 134400);     // 235200 B
  float*    score   = (float*)   (w + 369600);     //  33600 B
  int*      valid   = (int*)     (w + 403200);     //  33600 B
  int*      rank    = (int*)     (w + 436800);     //  33600 B
  float*    boxes_s = (float*)   (w + 470400);     // 134400 B
  int*      valid_s = (int*)     (w + 604800);     //  33600 B
  unsigned* mask    = (unsigned*)(w + 638400);     // 8400*263*4 = 8836800 B
  unsigned* removed = (unsigned*)(w + 9475200);    // 263*4 B  (total ~9.48 MB)

  k_prep<<<NBOX / 8, 256, 0, stream>>>(pred, boxes, det, score, valid);
  k_rank<<<(NBOX + 255) / 256, 256, 0, stream>>>(score, boxes, valid,
                                                 rank, boxes_s, valid_s);
  dim3 mg((NW + 31) / 32, (NBOX + 7) / 8);   // 9 x 1050 blocks
  k_mask<<<mg, 256, 0, stream>>>(boxes_s, valid_s, mask);
  k_collect<<<1, 1024, 0, stream>>>(mask, removed);
  k_final<<<(NBOX + 255) / 256, 256, 0, stream>>>(det, rank, valid, removed, out);
}